// PatchEmbedding_5523327943160
// MI455X (gfx1250) — compile-verified
//
#include <hip/hip_runtime.h>

typedef float v2f __attribute__((ext_vector_type(2)));
typedef float v8f __attribute__((ext_vector_type(8)));

#define PATCH     8
#define EMBED     256
#define N_MELS    128
#define N_FRAMES  2048
#define BATCHSZ   64
#define NUM_PATCHES 4096   // 16 mel-patch rows * 256 frame-patches
#define KDIM      64       // PATCH*PATCH

// One wave computes a 16-patch (M) x 64-embed (N) tile of the output with
// fp32 WMMA: 4 accumulator tiles of 16x16, K=64 -> 16 k-steps of
// v_wmma_f32_16x16x4_f32 per tile (64 WMMAs per wave, 4 independent chains).
__global__ __launch_bounds__(256) void patch_embed_wmma_f32(
    const float* __restrict__ x,     // (64,1,128,2048)
    const float* __restrict__ W,     // (256,64)
    const float* __restrict__ bias,  // (256,)
    const float* __restrict__ pos,   // (1,4096,256)
    float* __restrict__ out)         // (64,4096,256)
{
    const int wave = blockIdx.x * (blockDim.x >> 5) + (threadIdx.x >> 5);
    const int lane = threadIdx.x & 31;
    const int lh   = lane & 15;   // lane within half-wave
    const int hi   = lane >> 4;   // 0 = lanes 0-15, 1 = lanes 16-31

    const int et  = wave & 3;         // 64-embed group: 0..3
    const int pt  = wave >> 2;        // patch tile:     0..16383
    const int b   = pt >> 8;          // batch (256 patch-tiles per batch)
    const int t16 = pt & 255;         // tile within batch
    const int pm  = t16 >> 4;         // mel-patch row       0..15
    const int pf0 = (t16 & 15) << 4;  // first frame-patch   0..240 step 16
    const int n0  = (pm << 8) + pf0;  // first patch index within batch
    const int e0  = et << 6;          // first embed channel

    // ---- A tile: 16 patches x K=64, as 16 chunks of 16x4 (v2f per lane) ----
    // A layout (32-bit 16x4): VGPR0 = K[0]/K[2], VGPR1 = K[1]/K[3], M = lane%16.
    // k = r*8 + c inside an 8x8 patch; k0 is even, so (k0,k0+1) share a row.
    const float* xb = x + ((size_t)b * N_MELS + (size_t)pm * PATCH) * N_FRAMES
                        + (size_t)(pf0 + lh) * PATCH;
    v2f a[16];
#pragma unroll
    for (int kc = 0; kc < 16; ++kc) {
        const int k0 = kc * 4 + hi * 2;
        const int r  = k0 >> 3;
        const int c  = k0 & 7;
        const float* p = xb + r * N_FRAMES + c;
        a[kc] = v2f{ p[0], p[1] };   // global_load_b64
    }

    v8f acc[4];
#pragma unroll
    for (int t = 0; t < 4; ++t) acc[t] = v8f{};

    // ---- main K loop: 16 x (4 independent WMMA chains) ----
    // B layout (4x16): lanes 0-15 hold rows K=k0,k0+1; lanes 16-31 rows k0+2,k0+3;
    // N = lane%16.  B[k][e] = W[e][k] -> contiguous float2 from a W row.
#pragma unroll
    for (int kc = 0; kc < 16; ++kc) {
        const int k0 = kc * 4 + hi * 2;
#pragma unroll
        for (int t = 0; t < 4; ++t) {
            const float* wp = W + (size_t)(e0 + t * 16 + lh) * KDIM + k0;
            v2f bm = v2f{ wp[0], wp[1] };
            acc[t] = __builtin_amdgcn_wmma_f32_16x16x4_f32(
                false, a[kc], false, bm, (short)0, acc[t], false, false);
        }
    }

    // ---- epilogue: fuse + bias + pos_embed, coalesced stores ----
    // C/D layout: VGPR j -> M = j (lanes 0-15) or j+8 (lanes 16-31), N = lane%16.
#pragma unroll
    for (int t = 0; t < 4; ++t) {
        const int e  = e0 + t * 16 + lh;
        const float bv = bias[e];
#pragma unroll
        for (int j = 0; j < 8; ++j) {
            const int n = n0 + j + hi * 8;
            const float pv = pos[(size_t)n * EMBED + e];
            out[((size_t)b * NUM_PATCHES + n) * EMBED + e] = acc[t][j] + bv + pv;
        }
    }
}

extern "C" void kernel_launch(void* const* d_in, const int* in_sizes, int n_in,
                              void* d_out, int out_size, void* d_ws, size_t ws_size,
                              hipStream_t stream) {
    const float* x    = (const float*)d_in[0];  // (64,1,128,2048)
    const float* W    = (const float*)d_in[1];  // (256,64)
    const float* bias = (const float*)d_in[2];  // (256,)
    const float* pos  = (const float*)d_in[3];  // (1,4096,256)
    float* out = (float*)d_out;                 // (64,4096,256)

    // 64 batches * 256 patch-tiles * 4 embed-groups = 65536 waves
    // 8 waves (256 threads) per block -> 8192 blocks
    dim3 grid(8192), block(256);
    patch_embed_wmma_f32<<<grid, block, 0, stream>>>(x, W, bias, pos, out);
}